// ManifoldNet_46626164965583
// MI455X (gfx1250) — compile-verified
//
#include <hip/hip_runtime.h>
#include <hip/hip_bf16.h>

// ---------------------------------------------------------------------------
// ManifoldNet on MI455X (gfx1250, wave32).
// Dominant cost: stage-2 gram (512x512x480 per batch x16 ~= 4 GFLOP), run on
// the matrix pipe with V_WMMA_F32_16X16X4_F32 (f32 kept exact because the
// adjacency is only used for kNN ranking; bf16 would scramble neighbor order
// in 480-dim space). All other stages are tiny VALU work.
// ---------------------------------------------------------------------------

typedef __attribute__((ext_vector_type(2))) float v2f;
typedef __attribute__((ext_vector_type(8))) float v8f;

#define BB   16   // batch
#define NN   512  // points
#define DD   16   // manifold dim
#define KK   20   // neighbors
#define C1   30   // channels after mix1
#define C2   50   // channels after mix2
#define NC   40   // classes
#define F1   16   // stage-1 feature dim (D*1)
#define F2   480  // stage-2 feature dim (D*30)

// ---------------------------------------------------------------------------
// Softmax prep: wc1[20] = softmax(w_fm1[0,:]); wc2[c][k] = softmax_k(w_fm2[c,:])
// ---------------------------------------------------------------------------
__global__ __launch_bounds__(32) void prep_softmax_kernel(
    const float* __restrict__ w_fm1, const float* __restrict__ w_fm2,
    float* __restrict__ wc1, float* __restrict__ wc2) {
  int t = threadIdx.x;                 // rows 0..29 -> wc2, row 30 -> wc1
  if (t > C1) return;
  const float* src = (t == C1) ? w_fm1 : (w_fm2 + t * KK);
  float* dst       = (t == C1) ? wc1   : (wc2 + t * KK);
  float mx = -__builtin_inff();
  for (int k = 0; k < KK; ++k) mx = fmaxf(mx, src[k]);
  float s = 0.f;
  float e[KK];
  for (int k = 0; k < KK; ++k) { e[k] = __expf(src[k] - mx); s += e[k]; }
  float inv = 1.0f / s;
  for (int k = 0; k < KK; ++k) dst[k] = e[k] * inv;
}

// ---------------------------------------------------------------------------
// Row squared norms: sq[b*N+n] = sum_f X[b,n,f]^2
// ---------------------------------------------------------------------------
__global__ void row_norms_kernel(const float* __restrict__ X,
                                 float* __restrict__ sq, int F) {
  int row = blockIdx.x * blockDim.x + threadIdx.x;   // 0 .. B*N-1
  if (row >= BB * NN) return;
  const float* x = X + (size_t)row * F;
  float acc = 0.f;
  for (int f = 0; f < F; ++f) acc += x[f] * x[f];
  sq[row] = acc;
}

// ---------------------------------------------------------------------------
// adj[b][m][n] = sq[m] + sq[n] - 2 * (X_b X_b^T)[m][n] via WMMA f32 16x16x4.
// One wave per 16x16 output tile. For a gram the A and B tile loads share the
// same lane pattern (A layout: lanes 0-15 M=lane K={k,k+1}; lanes 16-31 same M,
// K={k+2,k+3}); C/D layout: VGPR r -> M = r + 8*(lane>=16), N = lane&15.
// ---------------------------------------------------------------------------
__global__ __launch_bounds__(32) void gram_adj_kernel(
    const float* __restrict__ X, const float* __restrict__ sq,
    float* __restrict__ adj, int F) {
  const int b    = blockIdx.z;
  const int ti   = blockIdx.y * 16;       // row tile base
  const int tj   = blockIdx.x * 16;       // col tile base
  const int lane = threadIdx.x;
  const int half = lane >> 4;             // 0: K pair {0,1}, 1: K pair {2,3}
  const int l16  = lane & 15;

  const float* Xb = X + (size_t)b * NN * F;
  const float* ra = Xb + (size_t)(ti + l16) * F;  // A row for this lane
  const float* rb = Xb + (size_t)(tj + l16) * F;  // B row for this lane

  v8f acc = {};
  for (int k = 0; k < F; k += 4) {
    const int ka = k + half * 2;
    v2f a, bmat;
    a.x    = ra[ka];
    a.y    = ra[ka + 1];
    bmat.x = rb[ka];
    bmat.y = rb[ka + 1];
    acc = __builtin_amdgcn_wmma_f32_16x16x4_f32(
        /*neg_a=*/false, a, /*neg_b=*/false, bmat,
        /*c_mod=*/(short)0, acc, /*reuse_a=*/false, /*reuse_b=*/false);
  }

  const float* sqb  = sq + b * NN;
  float*       adjb = adj + (size_t)b * NN * NN;
  const int    n    = tj + l16;
  const float  sqn  = sqb[n];
#pragma unroll
  for (int r = 0; r < 8; ++r) {
    const int m = ti + r + 8 * half;
    adjb[(size_t)m * NN + n] = sqb[m] + sqn - 2.0f * acc[r];
  }
}

// ---------------------------------------------------------------------------
// Top-K=20 smallest per row (self included), ascending with index tie-break,
// matching jax.lax.top_k(-adj). One wave per row: each lane caches 16 entries
// in registers; 20 rounds of lexicographic (dist,idx) wave-min via shfl_xor.
// ---------------------------------------------------------------------------
__global__ __launch_bounds__(32) void topk_kernel(
    const float* __restrict__ adj, int* __restrict__ idx) {
  const int row  = blockIdx.x;            // 0 .. B*N-1 (adj is [B*N, N] rows)
  const int lane = threadIdx.x;
  const float* a = adj + (size_t)row * NN;

  float v[16];
#pragma unroll
  for (int t = 0; t < 16; ++t) v[t] = a[lane * 16 + t];

  float lastD = -__builtin_inff();
  int   lastI = -1;
  for (int s = 0; s < KK; ++s) {
    float bd = __builtin_inff();
    int   bi = NN;
#pragma unroll
    for (int t = 0; t < 16; ++t) {
      const int   j = lane * 16 + t;
      const float d = v[t];
      const bool gtLast = (d > lastD) || (d == lastD && j > lastI);
      const bool ltBest = (d < bd)    || (d == bd    && j < bi);
      if (gtLast && ltBest) { bd = d; bi = j; }
    }
#pragma unroll
    for (int off = 16; off >= 1; off >>= 1) {
      const float od = __shfl_xor(bd, off, 32);
      const int   oi = __shfl_xor(bi, off, 32);
      if (od < bd || (od == bd && oi < bi)) { bd = od; bi = oi; }
    }
    if (lane == 0) idx[row * KK + s] = bi;
    lastD = bd;
    lastI = bi;
  }
}

// ---------------------------------------------------------------------------
// Stage-1 wFM + 1->30 mix, fused per point:
//   s[d]        = sum_k wc1[k] * x[b, idx[b,n,k], d]
//   fm1[bn][d*30+c] = s[d] * w_mix1[c]          (row-major [D, C1] flatten)
// ---------------------------------------------------------------------------
__global__ __launch_bounds__(64) void fm1_kernel(
    const float* __restrict__ x, const int* __restrict__ idx,
    const float* __restrict__ wc1, const float* __restrict__ w_mix1,
    float* __restrict__ fm1) {
  const int bn = blockIdx.x;              // b*N + n
  const int b  = bn / NN;
  const int t  = threadIdx.x;
  __shared__ float s[DD];
  __shared__ int   nb[KK];

  if (t < KK) nb[t] = idx[bn * KK + t];
  __syncthreads();
  if (t < DD) {
    float acc = 0.f;
    for (int k = 0; k < KK; ++k)
      acc += wc1[k] * x[((size_t)b * NN + nb[k]) * DD + t];
    s[t] = acc;
  }
  __syncthreads();
  for (int f = t; f < F2; f += 64)
    fm1[(size_t)bn * F2 + f] = s[f / C1] * w_mix1[f % C1];
}

// ---------------------------------------------------------------------------
// Stage-2 wFM + 30->50 mix + manifold distance + channel pool, fused per point:
//   fa[d*30+c]  = sum_k wc2[c][k] * fm1[b, idx2[b,n,k], d*30+c]
//   g[d][f]     = sum_c fa[d*30+c] * w_mix2[c][f]
//   dist[f]     = sqrt(sum_d (g[d][f]-m_last[d][f])^2 + 1e-8)
//   pooled[bn]  = mean_f dist[f]
// ---------------------------------------------------------------------------
__global__ __launch_bounds__(256) void fm2_last_kernel(
    const float* __restrict__ fm1, const int* __restrict__ idx2,
    const float* __restrict__ wc2, const float* __restrict__ w_mix2,
    const float* __restrict__ m_last, float* __restrict__ pooled) {
  const int bn = blockIdx.x;
  const int b  = bn / NN;
  const int t  = threadIdx.x;
  __shared__ float fa[F2];
  __shared__ float dist[C2];
  __shared__ int   nb[KK];

  if (t < KK) nb[t] = idx2[bn * KK + t];
  __syncthreads();
  // two passes cover 480 entries with 256 threads (all 256, then t<224)
  for (int f = t; f < F2; f += 256) {
    const int c = f % C1;
    float acc = 0.f;
    for (int k = 0; k < KK; ++k)
      acc += wc2[c * KK + k] * fm1[((size_t)b * NN + nb[k]) * F2 + f];
    fa[f] = acc;
  }
  __syncthreads();
  if (t < C2) {
    float ssum = 0.f;
    for (int d = 0; d < DD; ++d) {
      float g = 0.f;
#pragma unroll
      for (int c = 0; c < C1; ++c) g += fa[d * C1 + c] * w_mix2[c * C2 + t];
      const float df = g - m_last[d * C2 + t];
      ssum += df * df;
    }
    dist[t] = sqrtf(ssum + 1e-8f);
  }
  __syncthreads();
  if (t == 0) {
    float p = 0.f;
    for (int f = 0; f < C2; ++f) p += dist[f];
    pooled[bn] = p * (1.0f / C2);
  }
}

// ---------------------------------------------------------------------------
// out[b][cls] = b_cls[cls] + sum_n pooled[b][n] * w_cls[n][cls]
// ---------------------------------------------------------------------------
__global__ __launch_bounds__(64) void classify_kernel(
    const float* __restrict__ pooled, const float* __restrict__ w_cls,
    const float* __restrict__ b_cls, float* __restrict__ out) {
  const int b = blockIdx.x;
  const int t = threadIdx.x;
  if (t >= NC) return;
  float acc = b_cls[t];
  const float* p = pooled + b * NN;
  for (int n = 0; n < NN; ++n) acc += p[n] * w_cls[n * NC + t];
  out[b * NC + t] = acc;
}

// ---------------------------------------------------------------------------
extern "C" void kernel_launch(void* const* d_in, const int* in_sizes, int n_in,
                              void* d_out, int out_size, void* d_ws, size_t ws_size,
                              hipStream_t stream) {
  const float* inputs = (const float*)d_in[0];   // [16,512,16,1]
  const float* w_fm1  = (const float*)d_in[1];   // [1,20]
  const float* w_mix1 = (const float*)d_in[2];   // [1,30]
  const float* w_fm2  = (const float*)d_in[3];   // [30,20]
  const float* w_mix2 = (const float*)d_in[4];   // [30,50]
  const float* m_last = (const float*)d_in[5];   // [16,50]
  const float* w_cls  = (const float*)d_in[6];   // [512,40]
  const float* b_cls  = (const float*)d_in[7];   // [40]
  float* out = (float*)d_out;                    // [16,40]

  // Workspace layout (~33 MB total)
  float* adj    = (float*)d_ws;                              // B*N*N
  float* sq     = adj + (size_t)BB * NN * NN;                // B*N
  float* fm1    = sq + (size_t)BB * NN;                      // B*N*F2
  int*   idx    = (int*)(fm1 + (size_t)BB * NN * F2);        // B*N*K (reused)
  float* pooled = (float*)(idx + (size_t)BB * NN * KK);      // B*N
  float* wc1    = pooled + (size_t)BB * NN;                  // 20
  float* wc2    = wc1 + KK;                                  // 600

  const dim3 gramGrid(NN / 16, NN / 16, BB);

  // 0) softmax weights
  prep_softmax_kernel<<<1, 32, 0, stream>>>(w_fm1, w_fm2, wc1, wc2);

  // stage 1: adjacency on raw points (F=16), kNN, wFM + mix 1->30
  row_norms_kernel<<<(BB * NN + 127) / 128, 128, 0, stream>>>(inputs, sq, F1);
  gram_adj_kernel<<<gramGrid, 32, 0, stream>>>(inputs, sq, adj, F1);
  topk_kernel<<<BB * NN, 32, 0, stream>>>(adj, idx);
  fm1_kernel<<<BB * NN, 64, 0, stream>>>(inputs, idx, wc1, w_mix1, fm1);

  // stage 2: adjacency on features (F=480), kNN, wFM + mix 30->50 + pooling
  row_norms_kernel<<<(BB * NN + 127) / 128, 128, 0, stream>>>(fm1, sq, F2);
  gram_adj_kernel<<<gramGrid, 32, 0, stream>>>(fm1, sq, adj, F2);
  topk_kernel<<<BB * NN, 32, 0, stream>>>(adj, idx);
  fm2_last_kernel<<<BB * NN, 256, 0, stream>>>(fm1, idx, wc2, w_mix2, m_last, pooled);

  // classifier
  classify_kernel<<<BB, 64, 0, stream>>>(pooled, w_cls, b_cls, out);
}